// Deep_MT_DGF_GNN_32555852103876
// MI455X (gfx1250) — compile-verified
//
#include <hip/hip_runtime.h>
#include <hip/hip_bf16.h>
#include <math.h>

typedef float v2f __attribute__((ext_vector_type(2)));
typedef float v8f __attribute__((ext_vector_type(8)));
typedef int   v4i __attribute__((vector_size(4 * sizeof(int))));

#define BB 128
#define NN 256
#define HD 64

// ---- CDNA5 async global->LDS path (guarded; falls back to ds_store staging) ----
#ifndef TRY_ASYNC
#define TRY_ASYNC 1
#endif
#define HAVE_ASYNC_LDS 0
#if TRY_ASYNC && defined(__HIP_DEVICE_COMPILE__)
#  if defined(__has_builtin)
#    if __has_builtin(__builtin_amdgcn_global_load_async_to_lds_b128)
#      undef HAVE_ASYNC_LDS
#      define HAVE_ASYNC_LDS 1
#    endif
#  endif
#endif

#if HAVE_ASYNC_LDS
#  if __has_builtin(__builtin_amdgcn_s_wait_asynccnt)
#    define WAIT_ASYNC() __builtin_amdgcn_s_wait_asynccnt(0)
#  else
#    define WAIT_ASYNC() asm volatile("s_wait_asynccnt 0x0" ::: "memory")
#  endif
#else
#  define WAIT_ASYNC() do {} while (0)
#endif

#define LDS_STRIDE 68  // 64 + 4 pad: rows stay 16B aligned, conflict-free ds reads

__device__ __forceinline__ float elu1(float x) {
    return x > 0.f ? x : (__expf(x) - 1.f);
}

// Stage 64 rows x 64 floats (row stride gstride) into LDS tile [64][LDS_STRIDE].
__device__ __forceinline__ void stage64(const float* __restrict__ g0, int row0, int gstride,
                                        float* sdst, int tid) {
    #pragma unroll
    for (int it = 0; it < 4; ++it) {
        int grp = it * 256 + tid;          // 1024 groups of 4 floats
        int r = grp >> 4;
        int c4 = (grp & 15) << 2;
        const float* src = g0 + (size_t)(row0 + r) * gstride + c4;
        float* dst = sdst + r * LDS_STRIDE + c4;
#if HAVE_ASYNC_LDS
        v4i* sg = (v4i*)const_cast<float*>(src);
        auto gp = (__attribute__((address_space(1))) v4i*)sg;   // generic -> global
        auto lp = (__attribute__((address_space(3))) v4i*)dst;  // generic -> LDS
        __builtin_amdgcn_global_load_async_to_lds_b128(gp, lp, 0, 0);
#else
        const float4* s4 = (const float4*)src;
        float4 v = *s4;
        dst[0] = v.x; dst[1] = v.y; dst[2] = v.z; dst[3] = v.w;
#endif
    }
}

// ---------------- CNN branch ----------------

__global__ void conv3x3_k(const float* __restrict__ x, const float* __restrict__ w,
                          const float* __restrict__ bias, float* __restrict__ out,
                          int Cin, int Cout, int Hdim) {
    int total = BB * Cout * Hdim * Hdim;
    int idx = blockIdx.x * blockDim.x + threadIdx.x;
    if (idx >= total) return;
    int px = idx % Hdim; int t = idx / Hdim;
    int py = t % Hdim;   t /= Hdim;
    int c  = t % Cout;   int b = t / Cout;
    float acc = bias[c];
    const float* xb = x + (size_t)b * Cin * Hdim * Hdim;
    const float* wc = w + (size_t)c * Cin * 9;
    for (int ic = 0; ic < Cin; ++ic) {
        const float* xc = xb + ic * Hdim * Hdim;
        const float* wi = wc + ic * 9;
        for (int ky = 0; ky < 3; ++ky) {
            int yy = py + ky - 1;
            if ((unsigned)yy >= (unsigned)Hdim) continue;
            for (int kx = 0; kx < 3; ++kx) {
                int xx = px + kx - 1;
                if ((unsigned)xx >= (unsigned)Hdim) continue;
                acc += xc[yy * Hdim + xx] * wi[ky * 3 + kx];
            }
        }
    }
    out[idx] = acc;
}

__global__ void bnstats_k(const float* __restrict__ x, const float* __restrict__ g,
                          const float* __restrict__ beta, float* __restrict__ ss,
                          int C, int HW) {
    int c = blockIdx.x, tid = threadIdx.x;
    int n = BB * HW;
    float s = 0.f, s2 = 0.f;
    for (int t = tid; t < n; t += blockDim.x) {
        int b = t / HW, i = t - b * HW;
        float v = x[((size_t)b * C + c) * HW + i];
        s += v; s2 += v * v;
    }
    __shared__ float r1[256], r2[256];
    r1[tid] = s; r2[tid] = s2; __syncthreads();
    for (int st = 128; st > 0; st >>= 1) {
        if (tid < st) { r1[tid] += r1[tid + st]; r2[tid] += r2[tid + st]; }
        __syncthreads();
    }
    if (tid == 0) {
        float inv = 1.f / (float)n;
        float m = r1[0] * inv;
        float var = r2[0] * inv - m * m;
        float sc = g[c] * rsqrtf(var + 1e-5f);
        ss[c] = sc;
        ss[C + c] = beta[c] - m * sc;
    }
}

__global__ void bn_elu_maxpool_k(const float* __restrict__ x, const float* __restrict__ ss,
                                 float* __restrict__ out, int C, int Hin) {
    int Ho = Hin >> 1;
    int total = BB * C * Ho * Ho;
    int idx = blockIdx.x * blockDim.x + threadIdx.x;
    if (idx >= total) return;
    int px = idx % Ho; int t = idx / Ho;
    int py = t % Ho;   t /= Ho;
    int c  = t % C;    int b = t / C;
    float sc = ss[c], sh = ss[C + c];
    const float* xc = x + (((size_t)b * C + c) * Hin) * Hin;
    float m = -1e30f;
    for (int dy = 0; dy < 2; ++dy)
        for (int dx = 0; dx < 2; ++dx) {
            float v = sc * xc[(2 * py + dy) * Hin + 2 * px + dx] + sh;
            m = fmaxf(m, elu1(v));
        }
    out[idx] = m;
}

__global__ void bn_elu_avgpool_k(const float* __restrict__ x, const float* __restrict__ ss,
                                 float* __restrict__ feat) {
    int idx = blockIdx.x * blockDim.x + threadIdx.x;
    if (idx >= BB * 128) return;
    int q = idx % 4, c = (idx / 4) % 32, b = idx / 128;
    int qy = q >> 1, qx = q & 1;
    float sc = ss[c], sh = ss[32 + c];
    const float* xc = x + (((size_t)b * 32 + c) * 32) * 32;
    float s = 0.f;
    for (int y = 0; y < 16; ++y)
        for (int xk = 0; xk < 16; ++xk)
            s += elu1(sc * xc[(qy * 16 + y) * 32 + qx * 16 + xk] + sh);
    feat[idx] = s * (1.f / 256.f);
}

__global__ void lin_k(const float* __restrict__ in, const float* __restrict__ W,
                      const float* __restrict__ bias, float* __restrict__ out,
                      int rows, int idim, int odim, int ostride) {
    int idx = blockIdx.x * blockDim.x + threadIdx.x;
    if (idx >= rows * odim) return;
    int o = idx % odim, r = idx / odim;
    const float* ip = in + (size_t)r * idim;
    const float* wp = W + (size_t)o * idim;
    float acc = bias[o];
    for (int i = 0; i < idim; ++i) acc += ip[i] * wp[i];
    out[(size_t)r * ostride + o] = acc;
}

// ---------------- GNN branch (WMMA + LDS tiling) ----------------

__global__ void sq_k(const float* __restrict__ g, float* __restrict__ sq) {
    int idx = blockIdx.x * blockDim.x + threadIdx.x;
    if (idx >= BB * NN) return;
    const float* r = g + (size_t)idx * HD;
    float s = 0.f;
    #pragma unroll
    for (int d = 0; d < HD; ++d) s += r[d] * r[d];
    sq[idx] = s;
}

// Gram 64x64 region per 8-wave block; X row-blocks staged (async) in LDS; fused
// multi-head RBF epilogue in the WMMA accumulators -> adj[B,256,256].
__global__ void __launch_bounds__(256)
gram_adj_k(const float* __restrict__ X, const float* __restrict__ sq,
           const float* __restrict__ log_sig, int nheads, float* __restrict__ adj) {
    int b = blockIdx.z, tib = blockIdx.y, tjb = blockIdx.x;
    int tid = threadIdx.x;
    int lane = tid & 31, wave = tid >> 5;
    __shared__ float As[64][LDS_STRIDE];
    __shared__ float Bs[64][LDS_STRIDE];
    const float* Xb = X + (size_t)b * NN * HD;
    stage64(Xb, tib * 64, HD, &As[0][0], tid);
    stage64(Xb, tjb * 64, HD, &Bs[0][0], tid);
    WAIT_ASYNC();
    __syncthreads();

    float inv2s[8];
    for (int h = 0; h < nheads; ++h) {
        float s = __expf(log_sig[h]);
        inv2s[h] = 1.f / (2.f * s * s + 1e-6f);
    }
    float invH = 1.f / (float)nheads;

    int m = lane & 15;
    int ksel = (lane >> 4) << 1;          // lanes 0-15: K k,k+1 ; 16-31: K k+2,k+3
    const float* sqb = sq + b * NN;
    float* adjb = adj + (size_t)b * NN * NN;

    #pragma unroll
    for (int tt = 0; tt < 2; ++tt) {
        int t = wave + tt * 8;            // 16 sub-tiles (4x4) over 8 waves
        int si = t >> 2, sj = t & 3;
        const float* arow = &As[si * 16 + m][0];
        const float* brow = &Bs[sj * 16 + m][0];
        v8f acc = {};
        #pragma unroll
        for (int k = 0; k < HD; k += 4) {
            v2f a, bf;
            a.x  = arow[k + ksel]; a.y  = arow[k + ksel + 1];
            bf.x = brow[k + ksel]; bf.y = brow[k + ksel + 1];
            acc = __builtin_amdgcn_wmma_f32_16x16x4_f32(false, a, false, bf,
                                                        (short)0, acc, false, false);
        }
        int colg = tjb * 64 + sj * 16 + (lane & 15);
        int rbase = tib * 64 + si * 16 + ((lane >> 4) << 3);
        float sqc = sqb[colg];
        #pragma unroll
        for (int r = 0; r < 8; ++r) {
            int rowg = rbase + r;
            float d2 = fmaxf(sqb[rowg] + sqc - 2.f * acc[r], 0.f);
            float av = 0.f;
            for (int h = 0; h < nheads; ++h) av += __expf(-d2 * inv2s[h]);
            adjb[rowg * NN + colg] = av * invH;
        }
    }
}

// Y = adj @ X. 8-wave block computes 64 rows x all 64 cols; X staged per-64-K
// chunk in LDS (shared by all 8 waves); adj rows streamed from global once.
__global__ void __launch_bounds__(256)
adjx_k(const float* __restrict__ adj, const float* __restrict__ X, float* __restrict__ Y) {
    int b = blockIdx.z, tib = blockIdx.x;
    int tid = threadIdx.x;
    int lane = tid & 31, wave = tid >> 5;
    __shared__ float Xs[64][LDS_STRIDE];
    const float* Ab = adj + (size_t)b * NN * NN;
    const float* Xb = X + (size_t)b * NN * HD;
    int m = lane & 15;
    int ksel = (lane >> 4) << 1;
    int t0 = wave, t1 = wave + 8;
    int si0 = t0 >> 2, sj0 = t0 & 3;
    int si1 = t1 >> 2, sj1 = t1 & 3;
    const float* arow0 = Ab + (size_t)(tib * 64 + si0 * 16 + m) * NN;
    const float* arow1 = Ab + (size_t)(tib * 64 + si1 * 16 + m) * NN;
    v8f acc0 = {}, acc1 = {};
    for (int ch = 0; ch < 4; ++ch) {
        stage64(Xb, ch * 64, HD, &Xs[0][0], tid);
        WAIT_ASYNC();
        __syncthreads();
        int kb = ch * 64;
        #pragma unroll
        for (int k = 0; k < 64; k += 4) {
            int kk = k + ksel;
            v2f a0, a1, b0, b1;
            a0.x = arow0[kb + kk]; a0.y = arow0[kb + kk + 1];
            b0.x = Xs[kk][sj0 * 16 + m]; b0.y = Xs[kk + 1][sj0 * 16 + m];
            acc0 = __builtin_amdgcn_wmma_f32_16x16x4_f32(false, a0, false, b0,
                                                         (short)0, acc0, false, false);
            a1.x = arow1[kb + kk]; a1.y = arow1[kb + kk + 1];
            b1.x = Xs[kk][sj1 * 16 + m]; b1.y = Xs[kk + 1][sj1 * 16 + m];
            acc1 = __builtin_amdgcn_wmma_f32_16x16x4_f32(false, a1, false, b1,
                                                         (short)0, acc1, false, false);
        }
        __syncthreads();
    }
    float* Yb = Y + (size_t)b * NN * HD;
    int col0 = sj0 * 16 + (lane & 15), col1 = sj1 * 16 + (lane & 15);
    int rb0 = tib * 64 + si0 * 16 + ((lane >> 4) << 3);
    int rb1 = tib * 64 + si1 * 16 + ((lane >> 4) << 3);
    #pragma unroll
    for (int r = 0; r < 8; ++r) {
        Yb[(rb0 + r) * HD + col0] = acc0[r];
        Yb[(rb1 + r) * HD + col1] = acc1[r];
    }
}

// out = LN( elu(Y@W^T + bias) + Xin ), one 64-thread block per row
__global__ void proj_ln_k(const float* __restrict__ Y, const float* __restrict__ Xin,
                          const float* __restrict__ W, const float* __restrict__ bias,
                          const float* __restrict__ lng, const float* __restrict__ lnb,
                          float* __restrict__ out) {
    int row = blockIdx.x, o = threadIdx.x;
    const float* y = Y + (size_t)row * HD;
    float acc = bias[o];
    const float* wp = W + o * HD;
    #pragma unroll 8
    for (int d = 0; d < HD; ++d) acc += y[d] * wp[d];
    float t = elu1(acc) + Xin[(size_t)row * HD + o];
    __shared__ float r1[64], r2[64], ms[2];
    r1[o] = t; r2[o] = t * t; __syncthreads();
    for (int s = 32; s > 0; s >>= 1) {
        if (o < s) { r1[o] += r1[o + s]; r2[o] += r2[o + s]; }
        __syncthreads();
    }
    if (o == 0) {
        float mm = r1[0] * (1.f / 64.f);
        float var = r2[0] * (1.f / 64.f) - mm * mm;
        ms[0] = mm; ms[1] = rsqrtf(var + 1e-5f);
    }
    __syncthreads();
    out[(size_t)row * HD + o] = lng[o] * (t - ms[0]) * ms[1] + lnb[o];
}

__global__ void gmean_k(const float* __restrict__ g, float* __restrict__ comb) {
    int b = blockIdx.x, d = threadIdx.x;
    float s = 0.f;
    for (int n = 0; n < NN; ++n) s += g[((size_t)b * NN + n) * HD + d];
    comb[b * 192 + 64 + d] = s * (1.f / (float)NN);
}

// ---------------- Peripheral branch ----------------
__global__ void peri_k(const float* __restrict__ peri, const float* __restrict__ W,
                       const float* __restrict__ bias, const float* __restrict__ lng,
                       const float* __restrict__ lnb, float* __restrict__ comb) {
    int b = blockIdx.x, o = threadIdx.x;
    const float* p = peri + b * 55;
    float acc = bias[o];
    for (int i = 0; i < 55; ++i) acc += p[i] * W[o * 55 + i];
    __shared__ float r1[64], r2[64], ms[2];
    r1[o] = acc; r2[o] = acc * acc; __syncthreads();
    for (int s = 32; s > 0; s >>= 1) {
        if (o < s) { r1[o] += r1[o + s]; r2[o] += r2[o + s]; }
        __syncthreads();
    }
    if (o == 0) {
        float mm = r1[0] * (1.f / 64.f);
        float var = r2[0] * (1.f / 64.f) - mm * mm;
        ms[0] = mm; ms[1] = rsqrtf(var + 1e-5f);
    }
    __syncthreads();
    float ln = lng[o] * (acc - ms[0]) * ms[1] + lnb[o];
    comb[b * 192 + 128 + o] = elu1(ln);
}

// ---------------- Fusion DGF over 3 tokens (8 heads) ----------------
__global__ void fus_k(const float* __restrict__ comb, const float* __restrict__ log_sig,
                      const float* __restrict__ W, const float* __restrict__ bias,
                      const float* __restrict__ lng, const float* __restrict__ lnb,
                      float* __restrict__ fused) {
    int b = blockIdx.x, o = threadIdx.x;
    __shared__ float xs[3][64], ys[3][64], adj[9], sqv[3];
    __shared__ float r1[64], r2[64], ms[2];
    const float* cb = comb + b * 192;
    for (int t = 0; t < 3; ++t) xs[t][o] = cb[t * 64 + o];
    __syncthreads();
    if (o < 3) {
        float s = 0.f;
        for (int d = 0; d < 64; ++d) { float v = xs[o][d]; s += v * v; }
        sqv[o] = s;
    }
    __syncthreads();
    if (o < 9) {
        int i = o / 3, j = o % 3;
        float g = 0.f;
        for (int d = 0; d < 64; ++d) g += xs[i][d] * xs[j][d];
        float d2 = fmaxf(sqv[i] + sqv[j] - 2.f * g, 0.f);
        float av = 0.f;
        for (int h = 0; h < 8; ++h) {
            float s = __expf(log_sig[h]);
            av += __expf(-d2 / (2.f * s * s + 1e-6f));
        }
        adj[o] = av * (1.f / 8.f);
    }
    __syncthreads();
    for (int t = 0; t < 3; ++t)
        ys[t][o] = adj[t * 3] * xs[0][o] + adj[t * 3 + 1] * xs[1][o] + adj[t * 3 + 2] * xs[2][o];
    __syncthreads();
    for (int t = 0; t < 3; ++t) {
        float acc = bias[o];
        for (int d = 0; d < 64; ++d) acc += ys[t][d] * W[o * 64 + d];
        float v = elu1(acc) + xs[t][o];
        r1[o] = v; r2[o] = v * v; __syncthreads();
        for (int s = 32; s > 0; s >>= 1) {
            if (o < s) { r1[o] += r1[o + s]; r2[o] += r2[o + s]; }
            __syncthreads();
        }
        if (o == 0) {
            float mm = r1[0] * (1.f / 64.f);
            float var = r2[0] * (1.f / 64.f) - mm * mm;
            ms[0] = mm; ms[1] = rsqrtf(var + 1e-5f);
        }
        __syncthreads();
        fused[b * 192 + t * 64 + o] = lng[o] * (v - ms[0]) * ms[1] + lnb[o];
        __syncthreads();
    }
}

__global__ void head_k(const float* __restrict__ fused, const float* __restrict__ W1,
                       const float* __restrict__ b1, const float* __restrict__ W2,
                       const float* __restrict__ b2, float* __restrict__ out) {
    int b = blockIdx.x, j = threadIdx.x;
    const float* f = fused + b * 192;
    float acc = b1[j];
    for (int i = 0; i < 192; ++i) acc += f[i] * W1[j * 192 + i];
    __shared__ float h[64];
    h[j] = elu1(acc);
    __syncthreads();
    if (j < 2) {
        float s = b2[j];
        for (int i = 0; i < 64; ++i) s += h[i] * W2[j * 64 + i];
        out[b * 2 + j] = s;
    }
}

extern "C" void kernel_launch(void* const* d_in, const int* in_sizes, int n_in,
                              void* d_out, int out_size, void* d_ws, size_t ws_size,
                              hipStream_t stream) {
    (void)in_sizes; (void)n_in; (void)out_size; (void)ws_size;
    const float* maps  = (const float*)d_in[0];
    const float* stats = (const float*)d_in[1];
    const float* peri  = (const float*)d_in[2];
    const float* a1_w = (const float*)d_in[3];
    const float* a1_b = (const float*)d_in[4];
    const float* a2_w = (const float*)d_in[5];
    const float* a2_b = (const float*)d_in[6];
    const float* bn1_b = (const float*)d_in[7];
    const float* bn1_g = (const float*)d_in[8];
    const float* bn2_b = (const float*)d_in[9];
    const float* bn2_g = (const float*)d_in[10];
    const float* cnn_w = (const float*)d_in[11];
    const float* cnn_b = (const float*)d_in[12];
    const float* conv1_b = (const float*)d_in[13];
    const float* conv1_w = (const float*)d_in[14];
    const float* conv2_b = (const float*)d_in[15];
    const float* conv2_w = (const float*)d_in[16];
    // dgf layers: base 17 + 5*l : ln_b, ln_g, log_sig, proj_w, proj_b
    const float* fus_ln_b = (const float*)d_in[32];
    const float* fus_ln_g = (const float*)d_in[33];
    const float* fus_ls   = (const float*)d_in[34];
    const float* fus_w    = (const float*)d_in[35];
    const float* fus_b    = (const float*)d_in[36];
    const float* gnn_w = (const float*)d_in[37];
    const float* gnn_b = (const float*)d_in[38];
    const float* peri_w = (const float*)d_in[39];
    const float* peri_b = (const float*)d_in[40];
    const float* peri_ln_b = (const float*)d_in[41];
    const float* peri_ln_g = (const float*)d_in[42];
    const float* v1_w = (const float*)d_in[43];
    const float* v1_b = (const float*)d_in[44];
    const float* v2_w = (const float*)d_in[45];
    const float* v2_b = (const float*)d_in[46];

    float* out = (float*)d_out;
    char* w8 = (char*)d_ws;
    float* bn1_ss = (float*)(w8 + 0);         // 32 floats
    float* bn2_ss = (float*)(w8 + 256);       // 64 floats
    float* feat   = (float*)(w8 + 512);       // 128*128 floats
    float* comb   = (float*)(w8 + 66048);     // 128*3*64
    float* fusedf = (float*)(w8 + 164352);    // 128*3*64
    char*  big    = w8 + 262656;              // reused region (~59 MB peak)
    // CNN phase
    float* conv1r = (float*)(big);                  // 128*16*64*64
    float* pool1  = (float*)(big + 33554432);       // 128*16*32*32
    float* conv2r = (float*)(big + 41943040);       // 128*32*32*32
    // GNN phase (reuses the same region after CNN completes on the stream)
    float* gA     = (float*)(big);                  // 128*256*64
    float* gB     = (float*)(big + 8388608);        // 128*256*64
    float* adjbuf = (float*)(big + 16777216);       // 128*256*256
    float* ybuf   = (float*)(big + 50331648);       // 128*256*64
    float* sqbuf  = (float*)(big + 58720256);       // 128*256

    // ---- CNN branch ----
    {
        int tot1 = BB * 16 * 64 * 64;
        conv3x3_k<<<(tot1 + 255) / 256, 256, 0, stream>>>(maps, conv1_w, conv1_b, conv1r, 5, 16, 64);
        bnstats_k<<<16, 256, 0, stream>>>(conv1r, bn1_g, bn1_b, bn1_ss, 16, 64 * 64);
        int totp = BB * 16 * 32 * 32;
        bn_elu_maxpool_k<<<(totp + 255) / 256, 256, 0, stream>>>(conv1r, bn1_ss, pool1, 16, 64);
        int tot2 = BB * 32 * 32 * 32;
        conv3x3_k<<<(tot2 + 255) / 256, 256, 0, stream>>>(pool1, conv2_w, conv2_b, conv2r, 16, 32, 32);
        bnstats_k<<<32, 256, 0, stream>>>(conv2r, bn2_g, bn2_b, bn2_ss, 32, 32 * 32);
        bn_elu_avgpool_k<<<(BB * 128 + 255) / 256, 256, 0, stream>>>(conv2r, bn2_ss, feat);
        lin_k<<<(BB * 64 + 255) / 256, 256, 0, stream>>>(feat, cnn_w, cnn_b, comb, BB, 128, 64, 192);
    }

    // ---- GNN branch ----
    lin_k<<<(BB * NN * 64 + 255) / 256, 256, 0, stream>>>(stats, gnn_w, gnn_b, gA, BB * NN, 7, 64, 64);
    float* gin = gA;
    float* gout = gB;
    for (int l = 0; l < 3; ++l) {
        const float* ln_b  = (const float*)d_in[17 + 5 * l];
        const float* ln_g  = (const float*)d_in[18 + 5 * l];
        const float* lsig  = (const float*)d_in[19 + 5 * l];
        const float* pw    = (const float*)d_in[20 + 5 * l];
        const float* pb    = (const float*)d_in[21 + 5 * l];
        sq_k<<<(BB * NN + 255) / 256, 256, 0, stream>>>(gin, sqbuf);
        gram_adj_k<<<dim3(4, 4, BB), 256, 0, stream>>>(gin, sqbuf, lsig, 4, adjbuf);
        adjx_k<<<dim3(4, 1, BB), 256, 0, stream>>>(adjbuf, gin, ybuf);
        proj_ln_k<<<BB * NN, 64, 0, stream>>>(ybuf, gin, pw, pb, ln_g, ln_b, gout);
        float* tmp = gin; gin = gout; gout = tmp;
    }
    gmean_k<<<BB, 64, 0, stream>>>(gin, comb);

    // ---- Peripheral branch ----
    peri_k<<<BB, 64, 0, stream>>>(peri, peri_w, peri_b, peri_ln_g, peri_ln_b, comb);

    // ---- Fusion + heads ----
    fus_k<<<BB, 64, 0, stream>>>(comb, fus_ls, fus_w, fus_b, fus_ln_g, fus_ln_b, fusedf);
    head_k<<<BB, 64, 0, stream>>>(fusedf, v1_w, v1_b, v2_w, v2_b, out);           // v -> out[0:256]
    head_k<<<BB, 64, 0, stream>>>(fusedf, a1_w, a1_b, a2_w, a2_b, out + BB * 2);  // a -> out[256:512]
}